// NTMMemory_31190052504047
// MI455X (gfx1250) — compile-verified
//
#include <hip/hip_runtime.h>
#include <hip/hip_bf16.h>
#include <math.h>

typedef __attribute__((ext_vector_type(2))) float v2f;
typedef __attribute__((ext_vector_type(8))) float v8f;
typedef __attribute__((ext_vector_type(4))) float f4;
typedef int b128_t __attribute__((vector_size(16)));   // matches builtin param type

#ifndef __has_builtin
#define __has_builtin(x) 0
#endif

#if __has_builtin(__builtin_amdgcn_wmma_f32_16x16x4_f32)
#define HAVE_WMMA 1
#else
#define HAVE_WMMA 0
#endif

#if __has_builtin(__builtin_amdgcn_global_load_async_to_lds_b128) && \
    __has_builtin(__builtin_amdgcn_s_wait_asynccnt)
#define HAVE_ASYNC 1
#else
#define HAVE_ASYNC 0
#endif

#define Bc 64
#define Nc 2048
#define Mc 256
#define RPT 15            // 15 memory rows per tile; row 15 of the A-tile = k
#define TILES 137         // ceil(2048 / 15)
#define LROW 260          // 256 + 4 floats pad -> conflict-free ds_load_b64
#define NSPLIT 16
#define RPS (Nc / NSPLIT) // 128 rows per split in pass 3

// ---------------------------------------------------------------------------
// Pass 1: per (b, tile) wave computes cos[b, row] for 15 rows via a single
// Gram-matrix WMMA accumulation: A = [15 mem rows ; k], D = A * A^T.
//   D[i][i]  = sum of squares of row i  (norms; D[15][15] = ||k||^2)
//   D[i][15] = dot(row_i, k)            (num)
// ---------------------------------------------------------------------------
__global__ __launch_bounds__(32) void k1_cos(
    const float* __restrict__ mem, const float* __restrict__ kv,
    float* __restrict__ cosw)
{
  __shared__ float tl[16 * LROW];   // 16.25 KB tile
  __shared__ float red[32];         // [0..15]=num, [16..31]=sumsq

  const int b    = blockIdx.x;
  const int tile = blockIdx.y;      // 0..136
  const int lane = threadIdx.x;     // wave32
  const int row0 = tile * RPT;

  const float* memb = mem + (size_t)b * Nc * Mc;
  const float* kb   = kv + b * Mc;

  // ---- stage 15 memory rows (clamped at N-1) + k into LDS, 16B per lane ----
#pragma unroll
  for (int i = 0; i < 32; ++i) {
    int idx = i * 32 + lane;        // 1024 float4s total
    int r   = idx >> 6;             // 0..15
    int c   = (idx & 63) << 2;      // 0..252
    int rr  = row0 + r; if (rr > Nc - 1) rr = Nc - 1;
    const float* src = (r == 15) ? (kb + c) : (memb + (size_t)rr * Mc + c);
    float* dst = &tl[r * LROW + c];
#if HAVE_ASYNC
    __builtin_amdgcn_global_load_async_to_lds_b128(
        (__attribute__((address_space(1))) b128_t*)src,
        (__attribute__((address_space(3))) b128_t*)dst, 0, 0);
#else
    *(f4*)dst = *(const f4*)src;
#endif
  }
#if HAVE_ASYNC
  __builtin_amdgcn_s_wait_asynccnt(0);
#endif
  __syncthreads();

#if HAVE_WMMA
  // A-fragment (16x4 f32): lanes 0-15 hold row L, K={0,1}; lanes 16-31 K={2,3}.
  // B-fragment (4x16 f32) has the mirrored layout, so passing A as B gives A*A^T.
  const int ar = lane & 15;
  const int ao = (lane >> 4) << 1;  // 0 or 2
  v8f acc = {0.f, 0.f, 0.f, 0.f, 0.f, 0.f, 0.f, 0.f};
#pragma unroll 8
  for (int c4 = 0; c4 < 64; ++c4) {
    v2f a = *(const v2f*)&tl[ar * LROW + (c4 << 2) + ao];
    acc = __builtin_amdgcn_wmma_f32_16x16x4_f32(
        false, a, false, a, (short)0, acc, false, false);
  }
  // extract column 15 (num): lane15 holds rows 0..7, lane31 rows 8..15
  if (lane == 15) {
#pragma unroll
    for (int r = 0; r < 8; ++r) red[r] = acc[r];
  }
  if (lane == 31) {
#pragma unroll
    for (int r = 0; r < 8; ++r) red[8 + r] = acc[r];
  }
  // extract diagonal (sum of squares): (i,i) lives at vgpr i, lane i (i<8)
  // and vgpr i-8, lane i+16 (i>=8)
  float d = 0.f; int drow = -1;
#pragma unroll
  for (int r = 0; r < 8; ++r) {
    if (lane == r)      { d = acc[r]; drow = r; }
    if (lane == r + 24) { d = acc[r]; drow = r + 8; }
  }
  if (drow >= 0) red[16 + drow] = d;
#else
  // scalar fallback (keeps compile green if the builtin name probes wrong)
  if (lane < 16) {
    float nn = 0.f, ss = 0.f;
    for (int j = 0; j < Mc; ++j) {
      float x = tl[lane * LROW + j];
      nn += x * tl[15 * LROW + j];
      ss += x * x;
    }
    red[lane]      = nn;
    red[16 + lane] = ss;
  }
#endif
  __syncthreads();

  if (lane < RPT) {
    int row = row0 + lane;
    if (row < Nc) {
      float mn = fmaxf(sqrtf(red[16 + lane]), 1e-8f);
      float kn = fmaxf(sqrtf(red[16 + 15]),   1e-8f);
      cosw[b * Nc + row] = red[lane] / (mn * kn);
    }
  }
}

// ---------------------------------------------------------------------------
// Pass 2: per-batch softmax over N=2048 + fold in gating:
//   a[b,n] = w_r * (1 - w_lu_prev),  s[b] = sum_n w_r * w_w
// (sort / w_lu from the reference are dead code w.r.t. the output)
// ---------------------------------------------------------------------------
__global__ __launch_bounds__(256) void k2_softmax(
    const float* __restrict__ cosw, const float* __restrict__ wprev,
    const int* __restrict__ wlu, const float* __restrict__ g,
    float* __restrict__ aw, float* __restrict__ sw)
{
  __shared__ float sred[256];
  const int b = blockIdx.x, tid = threadIdx.x;
  const float* cb = cosw + b * Nc;

  float v[8];
  float mx = -3.0e38f;
#pragma unroll
  for (int i = 0; i < 8; ++i) { v[i] = cb[tid + (i << 8)]; mx = fmaxf(mx, v[i]); }
  sred[tid] = mx; __syncthreads();
  for (int s = 128; s > 0; s >>= 1) {
    if (tid < s) sred[tid] = fmaxf(sred[tid], sred[tid + s]);
    __syncthreads();
  }
  mx = sred[0]; __syncthreads();

  float sum = 0.f;
#pragma unroll
  for (int i = 0; i < 8; ++i) { v[i] = expf(v[i] - mx); sum += v[i]; }
  sred[tid] = sum; __syncthreads();
  for (int s = 128; s > 0; s >>= 1) {
    if (tid < s) sred[tid] += sred[tid + s];
    __syncthreads();
  }
  const float inv = 1.f / sred[0];
  __syncthreads();

  const float gb = g[b];
  float sp = 0.f;
#pragma unroll
  for (int i = 0; i < 8; ++i) {
    int n = tid + (i << 8);
    float wr  = v[i] * inv;
    float lu  = (float)wlu[b * Nc + n];
    float wrp = wprev[(size_t)(b * 2 + 1) * Nc + n];
    float ww  = gb * wrp + (1.f - gb) * lu;
    aw[b * Nc + n] = wr * (1.f - lu);
    sp += wr * ww;
  }
  sred[tid] = sp; __syncthreads();
  for (int s = 128; s > 0; s >>= 1) {
    if (tid < s) sred[tid] += sred[tid + s];
    __syncthreads();
  }
  if (tid == 0) sw[b] = sred[0];
}

// ---------------------------------------------------------------------------
// Pass 3: partial[split,b,m] = sum over 128 rows of a[b,n] * memory[b,n,m]
// (coalesced row-wise streaming; memory should still be L2-resident)
// ---------------------------------------------------------------------------
__global__ __launch_bounds__(256) void k3_partial(
    const float* __restrict__ mem, const float* __restrict__ aw,
    float* __restrict__ partial)
{
  __shared__ float al[RPS];
  const int b = blockIdx.x, sp = blockIdx.y, m = threadIdx.x;
  const int n0 = sp * RPS;
  if (m < RPS) al[m] = aw[b * Nc + n0 + m];
  __syncthreads();

  const float* base = mem + ((size_t)b * Nc + n0) * Mc + m;
  float acc = 0.f;
#pragma unroll 4
  for (int r = 0; r < RPS; ++r) acc = fmaf(al[r], base[(size_t)r * Mc], acc);
  partial[((size_t)sp * Bc + b) * Mc + m] = acc;
}

// ---------------------------------------------------------------------------
// Pass 4: out[b,m] = s[b]*k[b,m] + sum over splits (deterministic order)
// ---------------------------------------------------------------------------
__global__ __launch_bounds__(256) void k4_final(
    const float* __restrict__ partial, const float* __restrict__ sw,
    const float* __restrict__ kv, float* __restrict__ out)
{
  const int b = blockIdx.x, m = threadIdx.x;
  float acc = sw[b] * kv[b * Mc + m];
#pragma unroll
  for (int sp = 0; sp < NSPLIT; ++sp)
    acc += partial[((size_t)sp * Bc + b) * Mc + m];
  out[b * Mc + m] = acc;
}

extern "C" void kernel_launch(void* const* d_in, const int* in_sizes, int n_in,
                              void* d_out, int out_size, void* d_ws, size_t ws_size,
                              hipStream_t stream) {
  (void)in_sizes; (void)n_in; (void)out_size; (void)ws_size;
  const float* mem   = (const float*)d_in[0];   // (B,N,M)
  const float* kv    = (const float*)d_in[1];   // (B,M)
  const float* g     = (const float*)d_in[2];   // (B,1)
  // d_in[3] gamma, d_in[6] n: dead w.r.t. the returned `read`
  const float* wprev = (const float*)d_in[4];   // (B,2,N)
  const int*   wlu   = (const int*)d_in[5];     // (B,N)
  float* out = (float*)d_out;

  float* ws      = (float*)d_ws;
  float* cosw    = ws;                  // B*N
  float* aw      = cosw + Bc * Nc;      // B*N
  float* sw      = aw + Bc * Nc;        // B
  float* partial = sw + Bc;             // NSPLIT*B*M

  k1_cos    <<<dim3(Bc, TILES), 32, 0, stream>>>(mem, kv, cosw);
  k2_softmax<<<Bc, 256, 0, stream>>>(cosw, wprev, wlu, g, aw, sw);
  k3_partial<<<dim3(Bc, NSPLIT), 256, 0, stream>>>(mem, aw, partial);
  k4_final  <<<Bc, 256, 0, stream>>>(partial, sw, kv, out);
}